// GCN_34677565948888
// MI455X (gfx1250) — compile-verified
//
#include <hip/hip_runtime.h>

typedef __attribute__((ext_vector_type(2))) float v2f;
typedef __attribute__((ext_vector_type(8))) float v8f;

static constexpr int N  = 100000;
static constexpr int E  = 3200000;
static constexpr int D0 = 128;
static constexpr int D1 = 32;
static constexpr int D2 = 16;

// ---------------------------------------------------------------- degree ----
__global__ void init_deg(float* __restrict__ deg) {
  int i = blockIdx.x * blockDim.x + threadIdx.x;
  if (i < N) deg[i] = 1.0f;  // self-loop contributes 1 to every node's degree
}

__global__ void accum_deg(const long long* __restrict__ dst, float* __restrict__ deg) {
  int i = blockIdx.x * blockDim.x + threadIdx.x;
  if (i < E) atomicAdd(&deg[(int)dst[i]], 1.0f);
}

__global__ void make_dinv(float* __restrict__ d) {
  int i = blockIdx.x * blockDim.x + threadIdx.x;
  if (i < N) d[i] = rsqrtf(d[i]);  // deg >= 1 guaranteed by self-loops
}

// ------------------------------------------------------------- GEMM 1 -------
// h1[N,32] = x[N,128] @ W1[128,32] via V_WMMA_F32_16X16X4_F32.
// One wave computes a 16-row x 32-col tile (two 16x16 accumulators).
__global__ void __launch_bounds__(256)
gemm1(const float* __restrict__ x, const float* __restrict__ W1,
      float* __restrict__ h1) {
  __shared__ float sW[D0 * D1];  // 16 KB of the WGP's 320 KB LDS
  for (int i = threadIdx.x; i < D0 * D1; i += 256) sW[i] = W1[i];
  __syncthreads();

  const int wave = (blockIdx.x * blockDim.x + threadIdx.x) >> 5;
  const int lane = threadIdx.x & 31;
  if (wave >= N / 16) return;  // wave-uniform: EXEC stays all-ones for WMMA

  const int row = lane & 15;   // M for A, N for B/D
  const int kh  = lane >> 4;   // lanes 16-31 carry the upper K pair
  const float* __restrict__ xr = x + (long)(wave * 16 + row) * D0;

  v8f acc0 = {};
  v8f acc1 = {};
  for (int k = 0; k < D0; k += 4) {
    const int ka = k + kh * 2;
    v2f a;
    a.x = xr[ka];
    a.y = xr[ka + 1];
    v2f b0, b1;
    b0.x = sW[ka * D1 + row];
    b0.y = sW[(ka + 1) * D1 + row];
    b1.x = sW[ka * D1 + 16 + row];
    b1.y = sW[(ka + 1) * D1 + 16 + row];
    acc0 = __builtin_amdgcn_wmma_f32_16x16x4_f32(false, a, false, b0, (short)0,
                                                 acc0, false, false);
    acc1 = __builtin_amdgcn_wmma_f32_16x16x4_f32(false, a, false, b1, (short)0,
                                                 acc1, false, false);
  }

  // D layout: VGPR g -> M = g (lanes 0-15) / M = 8+g (lanes 16-31), N = lane&15
  const long mbase = (long)wave * 16 + kh * 8;
  for (int g = 0; g < 8; ++g) {
    const long r = (mbase + g) * D1;
    h1[r + row]      = acc0[g];
    h1[r + 16 + row] = acc1[g];
  }
}

// --------------------------------------------------------- edge scatter -----
__global__ void scatter1(const long long* __restrict__ ei,
                         const float* __restrict__ dinv,
                         const float* __restrict__ h1,
                         float* __restrict__ agg1) {
  long gid = (long)blockIdx.x * blockDim.x + threadIdx.x;
  if (gid >= (long)E * 8) return;
  const int e = (int)(gid >> 3);
  const int c = ((int)gid & 7) * 4;
  const int s = (int)ei[e];
  const int d = (int)ei[E + e];
  const float nrm = dinv[s] * dinv[d];
  const float4 m = *reinterpret_cast<const float4*>(h1 + (long)s * D1 + c);
  float* __restrict__ o = agg1 + (long)d * D1 + c;  // L2-resident target
  atomicAdd(o + 0, m.x * nrm);
  atomicAdd(o + 1, m.y * nrm);
  atomicAdd(o + 2, m.z * nrm);
  atomicAdd(o + 3, m.w * nrm);
}

// self-loop + bias + ReLU, in place on agg1
__global__ void post1(const float* __restrict__ h1, const float* __restrict__ dinv,
                      const float* __restrict__ b1, float* __restrict__ agg1) {
  long i = (long)blockIdx.x * blockDim.x + threadIdx.x;
  if (i >= (long)N * D1) return;
  const int n = (int)(i / D1);
  const int f = (int)(i % D1);
  const float di = dinv[n];
  const float v = agg1[i] + h1[i] * di * di + b1[f];
  agg1[i] = fmaxf(v, 0.0f);
}

// ------------------------------------------------------------- GEMM 2 -------
// h2[N,16] = relu_agg1[N,32] @ W2[32,16]; one 16x16 tile per wave, K=32.
__global__ void __launch_bounds__(256)
gemm2(const float* __restrict__ hin, const float* __restrict__ W2,
      float* __restrict__ h2) {
  __shared__ float sW[D1 * D2];  // 2 KB
  for (int i = threadIdx.x; i < D1 * D2; i += 256) sW[i] = W2[i];
  __syncthreads();

  const int wave = (blockIdx.x * blockDim.x + threadIdx.x) >> 5;
  const int lane = threadIdx.x & 31;
  if (wave >= N / 16) return;

  const int row = lane & 15;
  const int kh  = lane >> 4;
  const float* __restrict__ xr = hin + (long)(wave * 16 + row) * D1;

  v8f acc = {};
  for (int k = 0; k < D1; k += 4) {
    const int ka = k + kh * 2;
    v2f a;
    a.x = xr[ka];
    a.y = xr[ka + 1];
    v2f b;
    b.x = sW[ka * D2 + row];
    b.y = sW[(ka + 1) * D2 + row];
    acc = __builtin_amdgcn_wmma_f32_16x16x4_f32(false, a, false, b, (short)0,
                                                acc, false, false);
  }

  const long mbase = (long)wave * 16 + kh * 8;
  for (int g = 0; g < 8; ++g) h2[(mbase + g) * D2 + row] = acc[g];
}

__global__ void scatter2(const long long* __restrict__ ei,
                         const float* __restrict__ dinv,
                         const float* __restrict__ h2,
                         float* __restrict__ agg2) {
  long gid = (long)blockIdx.x * blockDim.x + threadIdx.x;
  if (gid >= (long)E * 4) return;
  const int e = (int)(gid >> 2);
  const int c = ((int)gid & 3) * 4;
  const int s = (int)ei[e];
  const int d = (int)ei[E + e];
  const float nrm = dinv[s] * dinv[d];
  const float4 m = *reinterpret_cast<const float4*>(h2 + (long)s * D2 + c);
  float* __restrict__ o = agg2 + (long)d * D2 + c;
  atomicAdd(o + 0, m.x * nrm);
  atomicAdd(o + 1, m.y * nrm);
  atomicAdd(o + 2, m.z * nrm);
  atomicAdd(o + 3, m.w * nrm);
}

// self-loop + bias + ReLU + final 16->1 projection, fused
__global__ void post2_final(const float* __restrict__ h2,
                            const float* __restrict__ agg2,
                            const float* __restrict__ dinv,
                            const float* __restrict__ b2,
                            const float* __restrict__ Wf,
                            const float* __restrict__ bf,
                            float* __restrict__ out) {
  int n = blockIdx.x * blockDim.x + threadIdx.x;
  if (n >= N) return;
  const float di = dinv[n];
  const float d2 = di * di;
  float s = 0.0f;
  for (int f = 0; f < D2; ++f) {
    const float v = agg2[(long)n * D2 + f] + h2[(long)n * D2 + f] * d2 + b2[f];
    s += fmaxf(v, 0.0f) * Wf[f];
  }
  out[n] = s + bf[0];
}

// ---------------------------------------------------------------- launch ----
extern "C" void kernel_launch(void* const* d_in, const int* in_sizes, int n_in,
                              void* d_out, int out_size, void* d_ws, size_t ws_size,
                              hipStream_t stream) {
  const float*     x  = (const float*)d_in[0];
  const long long* ei = (const long long*)d_in[1];  // int64 per reference
  const float*     W1 = (const float*)d_in[2];
  const float*     b1 = (const float*)d_in[3];
  const float*     W2 = (const float*)d_in[4];
  const float*     b2 = (const float*)d_in[5];
  const float*     Wf = (const float*)d_in[6];
  const float*     bf = (const float*)d_in[7];
  float* out = (float*)d_out;

  float* w    = (float*)d_ws;
  float* dinv = w;  w += N;
  float* h1   = w;  w += (long)N * D1;
  float* agg1 = w;  w += (long)N * D1;
  float* h2   = w;  w += (long)N * D2;
  float* agg2 = w;  w += (long)N * D2;   // total ~38.8 MB

  hipMemsetAsync(agg1, 0, sizeof(float) * (size_t)N * D1, stream);
  hipMemsetAsync(agg2, 0, sizeof(float) * (size_t)N * D2, stream);

  init_deg<<<(N + 255) / 256, 256, 0, stream>>>(dinv);
  accum_deg<<<(E + 255) / 256, 256, 0, stream>>>(ei + E, dinv);
  make_dinv<<<(N + 255) / 256, 256, 0, stream>>>(dinv);

  const int tiles = N / 16;                       // 6250 waves
  gemm1<<<(tiles + 7) / 8, 256, 0, stream>>>(x, W1, h1);

  scatter1<<<(int)(((long)E * 8 + 255) / 256), 256, 0, stream>>>(ei, dinv, h1, agg1);
  post1<<<(int)(((long)N * D1 + 255) / 256), 256, 0, stream>>>(h1, dinv, b1, agg1);

  gemm2<<<(tiles + 7) / 8, 256, 0, stream>>>(agg1, W2, h2);

  scatter2<<<(int)(((long)E * 4 + 255) / 256), 256, 0, stream>>>(ei, dinv, h2, agg2);
  post2_final<<<(N + 255) / 256, 256, 0, stream>>>(h2, agg2, dinv, b2, Wf, bf, out);
}